// RKMeansVectorQuantizer_54846732370495
// MI455X (gfx1250) — compile-verified
//
#include <hip/hip_runtime.h>
#include <hip/hip_bf16.h>

// ---------------------------------------------------------------------------
// VQ (k-means vector quantizer) forward for MI455X / gfx1250.
//   x:         [32768, 512] f32
//   centroids: [ 8192, 512] f32
// outputs (flat, in return order):
//   x_q  [32768*512] f32   (== gathered centroids, straight-through)
//   loss [1]         f32   (1.25 * mean((q - x)^2))
//   idx  [32768]     f32   (argmin indices, as float)
//
// Main kernel: fused exact-fp32 GEMM (v_wmma_f32_16x16x4_f32) + argmax of
// score(n,k) = x_n.c_k - 0.5*||c_k||^2, with double-buffered async
// global->LDS staging of centroid chunks (ASYNCcnt) overlapped with WMMA.
// ---------------------------------------------------------------------------

typedef __attribute__((ext_vector_type(2))) float v2f;
typedef __attribute__((ext_vector_type(8))) float v8f;
typedef __attribute__((ext_vector_type(4))) int   v4i;

#define VQ_N 32768
#define VQ_K 8192
#define VQ_D 512
#define MBLK 64    // rows of x per workgroup (4 waves x 16 rows)
#define NBLK 32    // centroids per chunk (each wave covers all 32 -> 2 tiles)
#define LDSS 516   // 512 + 4 pad -> all-64-bank conflict-free WMMA fetch

#if defined(__AMDGCN__) && __has_builtin(__builtin_amdgcn_global_load_async_to_lds_b128)
#define VQ_HAS_ASYNC 1
#else
#define VQ_HAS_ASYNC 0
#endif

__device__ __forceinline__ void vq_wait_async() {
#if VQ_HAS_ASYNC
#if __has_builtin(__builtin_amdgcn_s_wait_asynccnt)
    __builtin_amdgcn_s_wait_asynccnt(0);
#else
    asm volatile("s_wait_asynccnt 0x0" ::: "memory");
#endif
#endif
}

__device__ __forceinline__ void vq_copy_b128(const float* gsrc, float* ldst) {
#if VQ_HAS_ASYNC
    __builtin_amdgcn_global_load_async_to_lds_b128(
        (__attribute__((address_space(1))) v4i*)gsrc,
        (__attribute__((address_space(3))) v4i*)ldst, 0, 0);
#else
    *(float4*)ldst = *(const float4*)gsrc;
#endif
}

// ---------------------------------------------------------------- half norms
__global__ __launch_bounds__(256) void vq_cnorm(const float* __restrict__ c,
                                                float* __restrict__ cn) {
    const int lane = threadIdx.x & 31;
    const int wave = threadIdx.x >> 5;
    const int row  = blockIdx.x * 8 + wave;
    const float4* cr = (const float4*)(c + (size_t)row * VQ_D);
    float s = 0.f;
#pragma unroll
    for (int it = 0; it < 4; ++it) {
        float4 v = cr[lane + 32 * it];
        s += v.x * v.x + v.y * v.y + v.z * v.z + v.w * v.w;
    }
#pragma unroll
    for (int m = 16; m >= 1; m >>= 1) s += __shfl_xor(s, m, 32);
    if (lane == 0) cn[row] = 0.5f * s;
}

// ----------------------------------------------------- fused GEMM + argmax
__global__ __launch_bounds__(128) void vq_main(const float* __restrict__ x,
                                               const float* __restrict__ c,
                                               const float* __restrict__ cnorm,
                                               int* __restrict__ out_idx) {
    __shared__ float xs[MBLK][LDSS];       // 132,096 B
    __shared__ float cs[2][NBLK][LDSS];    // 132,096 B  (ping-pong)

    const int tid  = threadIdx.x;
    const int lane = tid & 31;
    const int wave = tid >> 5;             // 0..3, owns rows [16w, 16w+16)
    const int row_base = blockIdx.x * MBLK;

    // Stage the 64x512 x-block (once).
    for (int i = tid; i < MBLK * (VQ_D / 4); i += 128) {
        const int r  = i >> 7;             // VQ_D/4 == 128
        const int c4 = i & 127;
        vq_copy_b128(x + (size_t)(row_base + r) * VQ_D + c4 * 4,
                     &xs[r][c4 * 4]);
    }
    // Prefetch centroid chunk 0 into buffer 0.
    for (int i = tid; i < NBLK * (VQ_D / 4); i += 128) {
        const int r  = i >> 7;
        const int c4 = i & 127;
        vq_copy_b128(c + (size_t)r * VQ_D + c4 * 4, &cs[0][r][c4 * 4]);
    }
    vq_wait_async();
    __syncthreads();

    const v8f vzero = {};
    v8f acc0 = vzero, acc1 = vzero;
    float best[8];
    int   bidx[8];
#pragma unroll
    for (int j = 0; j < 8; ++j) { best[j] = -3.402823466e38f; bidx[j] = 0; }

    const int r16  = lane & 15;
    const int koff = (lane >> 4) * 2;      // K sub-offset per WMMA f32 layout

    int p = 0;
    for (int cb = 0; cb < VQ_K; cb += NBLK) {
        // Prefetch next chunk into the other buffer (overlaps with WMMA).
        if (cb + NBLK < VQ_K) {
            for (int i = tid; i < NBLK * (VQ_D / 4); i += 128) {
                const int r  = i >> 7;
                const int c4 = i & 127;
                vq_copy_b128(c + (size_t)(cb + NBLK + r) * VQ_D + c4 * 4,
                             &cs[p ^ 1][r][c4 * 4]);
            }
        }

        // 16 rows x 32 centroids per wave: two register-blocked 16x16 tiles.
        const float* ap  = &xs[wave * 16 + r16][koff];
        const float* bp0 = &cs[p][r16][koff];
        const float* bp1 = &cs[p][16 + r16][koff];
#pragma unroll 8
        for (int k0 = 0; k0 < VQ_D; k0 += 4) {
            v2f a  = *(const v2f*)(ap + k0);
            v2f b0 = *(const v2f*)(bp0 + k0);
            v2f b1 = *(const v2f*)(bp1 + k0);
            acc0 = __builtin_amdgcn_wmma_f32_16x16x4_f32(
                false, a, false, b0, (short)0, acc0, false, false);
            acc1 = __builtin_amdgcn_wmma_f32_16x16x4_f32(
                false, a, false, b1, (short)0, acc1, false, false);
        }

        const int col0 = cb + r16;
        const int col1 = col0 + 16;
        const float cn0 = cnorm[col0];
        const float cn1 = cnorm[col1];
#pragma unroll
        for (int j = 0; j < 8; ++j) {
            float s0 = acc0[j] - cn0;
            if (s0 > best[j]) { best[j] = s0; bidx[j] = col0; }
            float s1 = acc1[j] - cn1;
            if (s1 > best[j]) { best[j] = s1; bidx[j] = col1; }
        }
        acc0 = vzero;
        acc1 = vzero;

        vq_wait_async();   // my prefetch done (ASYNCcnt)
        __syncthreads();   // everyone's prefetch done, chunk p consumed
        p ^= 1;
    }

    // Each wave saw all centroids for its rows: reduce across the 16 lanes
    // holding the same output row; prefer lower index on ties (argmin-first).
#pragma unroll
    for (int j = 0; j < 8; ++j) {
        float b  = best[j];
        int   ix = bidx[j];
#pragma unroll
        for (int m = 1; m <= 8; m <<= 1) {
            float ob = __shfl_xor(b, m, 32);
            int   oi = __shfl_xor(ix, m, 32);
            if (ob > b || (ob == b && oi < ix)) { b = ob; ix = oi; }
        }
        if (r16 == 0) {
            const int rloc = wave * 16 + j + ((lane >> 4) << 3);
            out_idx[row_base + rloc] = ix;
        }
    }
}

// ------------------------------------------------- gather + loss accumulate
__global__ __launch_bounds__(128) void vq_gather(const float* __restrict__ x,
                                                 const float* __restrict__ c,
                                                 const int* __restrict__ idx,
                                                 float* __restrict__ out,
                                                 float* __restrict__ loss_acc) {
    __shared__ float partial[4];
    const int row = blockIdx.x;
    const int tid = threadIdx.x;
    const int k   = idx[row];
    float4 qv = ((const float4*)(c + (size_t)k * VQ_D))[tid];
    float4 xv = ((const float4*)(x + (size_t)row * VQ_D))[tid];
    ((float4*)(out + (size_t)row * VQ_D))[tid] = qv;
    const float dx = qv.x - xv.x, dy = qv.y - xv.y;
    const float dz = qv.z - xv.z, dw = qv.w - xv.w;
    float s = dx * dx + dy * dy + dz * dz + dw * dw;
#pragma unroll
    for (int m = 16; m >= 1; m >>= 1) s += __shfl_xor(s, m, 32);
    if ((tid & 31) == 0) partial[tid >> 5] = s;
    __syncthreads();
    if (tid == 0) {
        float t = partial[0] + partial[1] + partial[2] + partial[3];
        atomicAdd(loss_acc, t);
        out[(size_t)VQ_N * VQ_D + 1 + row] = (float)k;
    }
}

__global__ void vq_init(float* loss_acc) { *loss_acc = 0.f; }

__global__ void vq_finalize(const float* __restrict__ loss_acc,
                            float* __restrict__ out) {
    // commitment == codebook numerically -> loss = (1 + 0.25) * MSE
    out[(size_t)VQ_N * VQ_D] =
        loss_acc[0] * 1.25f / ((float)VQ_N * (float)VQ_D);
}

// ---------------------------------------------------------------------------
extern "C" void kernel_launch(void* const* d_in, const int* in_sizes, int n_in,
                              void* d_out, int out_size, void* d_ws,
                              size_t ws_size, hipStream_t stream) {
    const float* x = (const float*)d_in[0];
    const float* c = (const float*)d_in[1];
    float* out = (float*)d_out;

    char* ws = (char*)d_ws;
    float* cnorm    = (float*)ws;                              // 8192 f32
    int*   idxb     = (int*)(ws + (size_t)VQ_K * 4);           // 32768 i32
    float* loss_acc = (float*)(ws + (size_t)VQ_K * 4 + (size_t)VQ_N * 4);

    vq_init<<<1, 1, 0, stream>>>(loss_acc);
    vq_cnorm<<<VQ_K / 8, 256, 0, stream>>>(c, cnorm);
    vq_main<<<VQ_N / MBLK, 128, 0, stream>>>(x, c, cnorm, idxb);
    vq_gather<<<VQ_N, 128, 0, stream>>>(x, c, idxb, out, loss_acc);
    vq_finalize<<<1, 1, 0, stream>>>(loss_acc, out);
}